// ROIExpression_43696997270180
// MI455X (gfx1250) — compile-verified
//
#include <hip/hip_runtime.h>
#include <math.h>

// ---------------- problem constants (match reference) ----------------
constexpr int Nn  = 50000;   // nodes
constexpr int Ee  = 800000;  // edges (before self loops)
constexpr int Cc  = 128;     // channels
constexpr int Gg  = 16;      // graphs
constexpr int Fin = 256;     // input features
constexpr int Etot = Ee + Nn;                 // edges + self loops
constexpr float EPSF = 1e-5f;

typedef __attribute__((ext_vector_type(2))) float v2f;
typedef __attribute__((ext_vector_type(8))) float v8f;

// ---------------- helpers ----------------
__device__ __forceinline__ void atomicMaxF(float* addr, float v) {
  // sign-split trick: valid with -inf init
  if (v >= 0.f) atomicMax((int*)addr, __float_as_int(v));
  else          atomicMin((unsigned int*)addr, __float_as_uint(v));
}

__device__ __forceinline__ float waveReduceSum(float v) {
#pragma unroll
  for (int off = 16; off >= 1; off >>= 1) v += __shfl_xor(v, off, 32);
  return v;
}

// ---------------- WMMA fp32 GEMM: out[M,128] = A[M,K] @ W[128,K]^T + bias ----------------
// One wave computes a 16x64 strip: 4 accumulator frags, A frag reused 4x.
__global__ __launch_bounds__(256) void gemm_nt_wmma_kernel(
    const float* __restrict__ A, const float* __restrict__ W,
    const float* __restrict__ bias, float* __restrict__ out,
    int M, int K)
{
  const int lane = threadIdx.x;          // 0..31
  const int half = lane >> 4;            // 0 | 1
  const int l16  = lane & 15;
  const int rowTile = blockIdx.x * 8 + threadIdx.y;
  if (rowTile * 16 >= M) return;         // wave-uniform exit: EXEC all-ones at WMMA
  const int row0 = rowTile * 16;
  const int col0 = blockIdx.y * 64;

  v8f acc[4] = {};

  // A 16x4 f32 frag layout (ISA 7.12.2): lanes 0-15 hold (M=l16, K=k0,k0+1),
  // lanes 16-31 hold (M=l16, K=k0+2,k0+3) -> per-lane base offset half*2.
  const float* ap = A + (size_t)(row0 + l16) * K + half * 2;
  // B 4x16 frag mirrored: lane half selects K pair, l16 selects N column.
  const float* wp = W + (size_t)(col0 + l16) * K + half * 2;

#pragma unroll 2
  for (int k0 = 0; k0 < K; k0 += 4) {
    v2f afr = *(const v2f*)(ap + k0);
#pragma unroll
    for (int t = 0; t < 4; ++t) {
      v2f bfr = *(const v2f*)(wp + (size_t)t * 16 * K + k0);
      acc[t] = __builtin_amdgcn_wmma_f32_16x16x4_f32(
          /*neg_a=*/false, afr, /*neg_b=*/false, bfr,
          /*c_mod=*/(short)0, acc[t], /*reuse_a=*/false, /*reuse_b=*/false);
    }
  }

  // D layout: vgpr r -> row r (lanes 0-15) / row r+8 (lanes 16-31), col = l16
#pragma unroll
  for (int t = 0; t < 4; ++t) {
    const int col = col0 + t * 16 + l16;
    const float bv = bias ? bias[col] : 0.f;
#pragma unroll
    for (int r = 0; r < 8; ++r) {
      const int row = row0 + r + 8 * half;
      out[(size_t)row * Cc + col] = acc[t][r] + bv;
    }
  }
}

// ---------------- per-row LayerNorm + LeakyReLU(0.01) (embed LN) ----------------
__global__ __launch_bounds__(256) void rowln_leaky_kernel(
    const float* __restrict__ in, const float* __restrict__ w,
    const float* __restrict__ b, float* __restrict__ out, int M)
{
  const int row = blockIdx.x * 8 + threadIdx.y;
  if (row >= M) return;
  const int lane = threadIdx.x;
  float4 v = ((const float4*)(in + (size_t)row * Cc))[lane];
  float s = waveReduceSum(v.x + v.y + v.z + v.w);
  const float mean = s * (1.f / Cc);
  float4 d = {v.x - mean, v.y - mean, v.z - mean, v.w - mean};
  float q = waveReduceSum(d.x*d.x + d.y*d.y + d.z*d.z + d.w*d.w);
  const float inv = rsqrtf(q * (1.f / Cc) + EPSF);
  float4 wv = ((const float4*)w)[lane];
  float4 bv = ((const float4*)b)[lane];
  float4 y = {d.x*inv*wv.x + bv.x, d.y*inv*wv.y + bv.y,
              d.z*inv*wv.z + bv.z, d.w*inv*wv.w + bv.w};
  y.x = y.x > 0.f ? y.x : 0.01f * y.x;
  y.y = y.y > 0.f ? y.y : 0.01f * y.y;
  y.z = y.z > 0.f ? y.z : 0.01f * y.z;
  y.w = y.w > 0.f ? y.w : 0.01f * y.w;
  ((float4*)(out + (size_t)row * Cc))[lane] = y;
}

// ---------------- init kernels ----------------
__global__ void init_small_kernel(float* out, double* gsum, double* bnacc) {
  const int i = blockIdx.x * blockDim.x + threadIdx.x;
  if (i < Gg * Cc) out[i] = 0.f;
  if (i < 4) gsum[i] = 0.0;
  if (i < 2 * Cc) bnacc[i] = 0.0;
}

__global__ __launch_bounds__(256) void init_layer_kernel(
    float* acc, float* mmax, float* ssum) {
  const size_t idx = (size_t)blockIdx.x * blockDim.x + threadIdx.x;
  if (idx < (size_t)Nn * Cc) acc[idx] = 0.f;
  if (idx < Nn) { mmax[idx] = -HUGE_VALF; ssum[idx] = 0.f; }
}

// ---------------- edge pass 1: attention logits + segment max ----------------
__global__ __launch_bounds__(256) void edge_e_kernel(
    const float* __restrict__ xl, const float* __restrict__ xr,
    const int* __restrict__ ei, const float* __restrict__ eattr,
    const float* __restrict__ We, const float* __restrict__ att,
    float* __restrict__ evals, float* __restrict__ mmax)
{
  const int j = blockIdx.x * 8 + threadIdx.y;
  if (j >= Etot) return;
  const int lane = threadIdx.x;
  int src, dst; float ea;
  if (j < Ee) { src = ei[j]; dst = ei[Ee + j]; ea = eattr[j]; }
  else        { src = j - Ee; dst = src; ea = 0.f; }
  float4 l = ((const float4*)(xl + (size_t)src * Cc))[lane];
  float4 r = ((const float4*)(xr + (size_t)dst * Cc))[lane];
  float4 w = ((const float4*)We)[lane];
  float4 a = ((const float4*)att)[lane];
  float e = 0.f, z;
  z = l.x + r.x + ea * w.x; z = z > 0.f ? z : 0.2f * z; e += z * a.x;
  z = l.y + r.y + ea * w.y; z = z > 0.f ? z : 0.2f * z; e += z * a.y;
  z = l.z + r.z + ea * w.z; z = z > 0.f ? z : 0.2f * z; e += z * a.z;
  z = l.w + r.w + ea * w.w; z = z > 0.f ? z : 0.2f * z; e += z * a.w;
  e = waveReduceSum(e);
  if (lane == 0) { evals[j] = e; atomicMaxF(&mmax[dst], e); }
}

// ---------------- edge pass 2: exp, denom, weighted scatter-add ----------------
__global__ __launch_bounds__(256) void edge_agg_kernel(
    const float* __restrict__ xl, const int* __restrict__ ei,
    const float* __restrict__ evals, const float* __restrict__ mmax,
    float* __restrict__ ssum, float* __restrict__ acc)
{
  const int j = blockIdx.x * 8 + threadIdx.y;
  if (j >= Etot) return;
  const int lane = threadIdx.x;
  int src, dst;
  if (j < Ee) { src = ei[j]; dst = ei[Ee + j]; }
  else        { src = j - Ee; dst = src; }
  const float a = expf(evals[j] - mmax[dst]);
  if (lane == 0) atomicAdd(&ssum[dst], a);
  float4 l = ((const float4*)(xl + (size_t)src * Cc))[lane];
  float* arow = acc + (size_t)dst * Cc + lane * 4;
  atomicAdd(arow + 0, a * l.x);
  atomicAdd(arow + 1, a * l.y);
  atomicAdd(arow + 2, a * l.z);
  atomicAdd(arow + 3, a * l.w);
}

// ---------------- finalize node: divide by denom, + bias, global stats (fp64) ----------------
__global__ __launch_bounds__(256) void node_finish_kernel(
    float* __restrict__ acc, const float* __restrict__ ssum,
    const float* __restrict__ bias, double* __restrict__ gs)
{
  __shared__ float red[2][8];
  float lsum = 0.f, lsq = 0.f;
  for (size_t idx = (size_t)blockIdx.x * 256 + threadIdx.x;
       idx < (size_t)Nn * Cc; idx += (size_t)gridDim.x * 256) {
    const int n = (int)(idx >> 7);
    const int c = (int)(idx & 127);
    const float v = acc[idx] / ssum[n] + bias[c];
    acc[idx] = v;
    lsum += v; lsq += v * v;
  }
  lsum = waveReduceSum(lsum);
  lsq  = waveReduceSum(lsq);
  const int lane = threadIdx.x & 31, wid = threadIdx.x >> 5;
  if (lane == 0) { red[0][wid] = lsum; red[1][wid] = lsq; }
  __syncthreads();
  if (threadIdx.x == 0) {
    double s0 = 0.0, s1 = 0.0;
#pragma unroll
    for (int i = 0; i < 8; ++i) { s0 += red[0][i]; s1 += red[1][i]; }
    atomicAdd(gs, s0);
    atomicAdd(gs + 1, s1);
  }
}

// ---------------- graph-mode LayerNorm + LeakyReLU(0.01) ----------------
__global__ __launch_bounds__(256) void graph_ln_leaky_kernel(
    const float* __restrict__ acc, const double* __restrict__ gs,
    const float* __restrict__ w, const float* __restrict__ b,
    float* __restrict__ out)
{
  const size_t idx = (size_t)blockIdx.x * blockDim.x + threadIdx.x;
  if (idx >= (size_t)Nn * Cc) return;
  const double invn = 1.0 / ((double)Nn * Cc);
  const double mu = gs[0] * invn;
  const double var = gs[1] * invn - mu * mu;
  const float inv = rsqrtf((float)var + EPSF);
  const int c = (int)(idx & 127);
  float y = (acc[idx] - (float)mu) * inv * w[c] + b[c];
  out[idx] = y > 0.f ? y : 0.01f * y;
}

// ---------------- JumpingKnowledge max ----------------
__global__ __launch_bounds__(256) void jk_max_kernel(
    const float* __restrict__ h0, const float* __restrict__ h1,
    const float* __restrict__ h2, float* __restrict__ out)
{
  const size_t idx = (size_t)blockIdx.x * blockDim.x + threadIdx.x;
  if (idx >= (size_t)Nn * Cc) return;
  out[idx] = fmaxf(fmaxf(h0[idx], h1[idx]), h2[idx]);
}

// ---------------- BatchNorm column stats (fp64 accum) ----------------
__global__ __launch_bounds__(128) void bn_stats_kernel(
    const float* __restrict__ p, double* __restrict__ bnacc)
{
  const int c = threadIdx.x;
  const int r0 = blockIdx.x * 256;
  const int r1 = min(r0 + 256, Nn);
  double s = 0.0, q = 0.0;
  for (int r = r0; r < r1; ++r) {
    const float v = p[(size_t)r * Cc + c];
    s += v; q += (double)v * v;
  }
  atomicAdd(&bnacc[c], s);
  atomicAdd(&bnacc[Cc + c], q);
}

// ---------------- BatchNorm apply + ReLU ----------------
__global__ __launch_bounds__(256) void bn_relu_kernel(
    const float* __restrict__ p, const double* __restrict__ bnacc,
    const float* __restrict__ w, const float* __restrict__ b,
    float* __restrict__ out)
{
  const size_t idx = (size_t)blockIdx.x * blockDim.x + threadIdx.x;
  if (idx >= (size_t)Nn * Cc) return;
  const int c = (int)(idx & 127);
  const double mu = bnacc[c] / Nn;
  const double var = bnacc[Cc + c] / Nn - mu * mu;
  const float inv = rsqrtf((float)var + EPSF);
  const float y = (p[idx] - (float)mu) * inv * w[c] + b[c];
  out[idx] = y > 0.f ? y : 0.f;
}

// ---------------- global_add_pool of |pred| per graph ----------------
__global__ __launch_bounds__(128) void pool_kernel(
    const float* __restrict__ pred, const int* __restrict__ batch,
    float* __restrict__ out)
{
  __shared__ float lacc[Gg * Cc];
  const int c = threadIdx.x;
#pragma unroll
  for (int g = 0; g < Gg; ++g) lacc[g * Cc + c] = 0.f;  // column-exclusive, no sync needed
  const int r0 = blockIdx.x * 256;
  const int r1 = min(r0 + 256, Nn);
  for (int r = r0; r < r1; ++r) {
    const int g = batch[r];
    lacc[g * Cc + c] += fabsf(pred[(size_t)r * Cc + c]);
  }
#pragma unroll
  for (int g = 0; g < Gg; ++g) atomicAdd(&out[g * Cc + c], lacc[g * Cc + c]);
}

// ---------------- host orchestration ----------------
extern "C" void kernel_launch(void* const* d_in, const int* in_sizes, int n_in,
                              void* d_out, int out_size, void* d_ws, size_t ws_size,
                              hipStream_t stream) {
  (void)in_sizes; (void)n_in; (void)out_size; (void)ws_size;
  const float* x        = (const float*)d_in[0];
  const int*   ei       = (const int*)d_in[1];
  const float* eattr    = (const float*)d_in[2];
  const int*   batch    = (const int*)d_in[3];
  const float* emb_W    = (const float*)d_in[4];
  const float* emb_b    = (const float*)d_in[5];
  const float* emb_ln_w = (const float*)d_in[6];
  const float* emb_ln_b = (const float*)d_in[7];
  const float* gWl[2]   = {(const float*)d_in[8],  (const float*)d_in[17]};
  const float* gbl[2]   = {(const float*)d_in[9],  (const float*)d_in[18]};
  const float* gWr[2]   = {(const float*)d_in[10], (const float*)d_in[19]};
  const float* gbr[2]   = {(const float*)d_in[11], (const float*)d_in[20]};
  const float* gWe[2]   = {(const float*)d_in[12], (const float*)d_in[21]};
  const float* gatt[2]  = {(const float*)d_in[13], (const float*)d_in[22]};
  const float* gbias[2] = {(const float*)d_in[14], (const float*)d_in[23]};
  const float* glnw[2]  = {(const float*)d_in[15], (const float*)d_in[24]};
  const float* glnb[2]  = {(const float*)d_in[16], (const float*)d_in[25]};
  const float* proj_W0  = (const float*)d_in[26];
  const float* proj_b0  = (const float*)d_in[27];
  const float* bn_w     = (const float*)d_in[28];
  const float* bn_b     = (const float*)d_in[29];
  const float* proj_W1  = (const float*)d_in[30];

  // workspace carve (aligned 256B)
  char* w = (char*)d_ws;
  auto carve = [&](size_t bytes) -> void* {
    void* p = (void*)w;
    w += (bytes + 255) & ~(size_t)255;
    return p;
  };
  const size_t NCb = (size_t)Nn * Cc * sizeof(float);
  float* h0    = (float*)carve(NCb);
  float* h1    = (float*)carve(NCb);
  float* h2    = (float*)carve(NCb);
  float* xl    = (float*)carve(NCb);              // also hjk
  float* xr    = (float*)carve(NCb);              // also q (post-BN)
  float* acc   = (float*)carve(NCb);              // GEMM staging / aggregation
  float* evals = (float*)carve((size_t)Etot * sizeof(float));
  float* mmax  = (float*)carve((size_t)Nn * sizeof(float));
  float* ssum  = (float*)carve((size_t)Nn * sizeof(float));
  double* gsum = (double*)carve(4 * sizeof(double));        // [layer0 S1,S2, layer1 S1,S2]
  double* bnacc= (double*)carve(2 * Cc * sizeof(double));   // col sum, col sumsq

  const dim3 blk256(32, 8);
  const int ewGrid = (Etot + 7) / 8;
  const int ncGrid = (Nn * Cc + 255) / 256;
  const int rowGrid = (Nn / 256) + 1;  // strips of 256 rows -> 196 blocks

  auto gemm = [&](const float* A, const float* Wm, const float* bias,
                  float* out, int M, int K) {
    dim3 grid((M / 16 + 7) / 8, Cc / 64);
    gemm_nt_wmma_kernel<<<grid, blk256, 0, stream>>>(A, Wm, bias, out, M, K);
  };

  // 0) zero output + fp64 accumulators
  init_small_kernel<<<8, 256, 0, stream>>>((float*)d_out, gsum, bnacc);

  // 1) node embed: GEMM -> row LN -> leaky -> h0
  gemm(x, emb_W, emb_b, acc, Nn, Fin);
  rowln_leaky_kernel<<<(Nn + 7) / 8, blk256, 0, stream>>>(acc, emb_ln_w, emb_ln_b, h0, Nn);

  // 2) two GATv2 + graph-LN + leaky blocks
  const float* hin = h0;
  float* houts[2] = {h1, h2};
  for (int li = 0; li < 2; ++li) {
    init_layer_kernel<<<ncGrid, 256, 0, stream>>>(acc, mmax, ssum);
    gemm(hin, gWl[li], gbl[li], xl, Nn, Cc);
    gemm(hin, gWr[li], gbr[li], xr, Nn, Cc);
    edge_e_kernel<<<ewGrid, blk256, 0, stream>>>(xl, xr, ei, eattr, gWe[li], gatt[li], evals, mmax);
    edge_agg_kernel<<<ewGrid, blk256, 0, stream>>>(xl, ei, evals, mmax, ssum, acc);
    node_finish_kernel<<<4096, 256, 0, stream>>>(acc, ssum, gbias[li], gsum + 2 * li);
    graph_ln_leaky_kernel<<<ncGrid, 256, 0, stream>>>(acc, gsum + 2 * li, glnw[li], glnb[li], houts[li]);
    hin = houts[li];
  }

  // 3) JK max -> xl (hjk)
  jk_max_kernel<<<ncGrid, 256, 0, stream>>>(h0, h1, h2, xl);

  // 4) projection head
  gemm(xl, proj_W0, proj_b0, acc, Nn, Cc);                       // p
  bn_stats_kernel<<<rowGrid, 128, 0, stream>>>(acc, bnacc);
  bn_relu_kernel<<<ncGrid, 256, 0, stream>>>(acc, bnacc, bn_w, bn_b, xr);  // q
  gemm(xr, proj_W1, nullptr, acc, Nn, Cc);                       // pred

  // 5) pooled |pred| per graph
  pool_kernel<<<rowGrid, 128, 0, stream>>>(acc, batch, (float*)d_out);
}